// ODEFuncGread_17497696764519
// MI455X (gfx1250) — compile-verified
//
#include <hip/hip_runtime.h>
#include <math.h>

// ---------------------------------------------------------------------------
// ODEFuncGread: out = alpha*(A@x - x) + beta*(-(x-1)*x) + 0.1*clip(s*x0,-1,1)
// Memory-bound sparse scatter: E=1.6M edges, D=128, x/out L2-resident (51MB).
// CDNA5 path used: GLOBAL_LOAD_ASYNC_TO_LDS_B32 edge streaming (ASYNCcnt
// double buffering) + hardware global_atomic_add_f32 scatter.
// ---------------------------------------------------------------------------

#define TILE 256   // edges staged per block iteration
#define DFEAT 128  // feature dim (reference constant)

// ---- async global->LDS support (gfx1250), with compile-safe fallback ------
#if defined(__has_builtin)
#if __has_builtin(__builtin_amdgcn_global_load_async_to_lds_b32)
#define HAS_ASYNC_LDS 1
#endif
#endif
#ifndef HAS_ASYNC_LDS
#define HAS_ASYNC_LDS 0
#endif

#if HAS_ASYNC_LDS
typedef __attribute__((address_space(1))) int* gint_p;
typedef __attribute__((address_space(3))) int* lint_p;
__device__ __forceinline__ void async_copy_b32(const void* g, void* l) {
  __builtin_amdgcn_global_load_async_to_lds_b32((gint_p)g, (lint_p)l, 0, 0);
}
#endif

__device__ __forceinline__ void wait_async_le(int keep3) {
#if defined(__has_builtin) && __has_builtin(__builtin_amdgcn_s_wait_asynccnt)
  if (keep3) __builtin_amdgcn_s_wait_asynccnt(3);
  else       __builtin_amdgcn_s_wait_asynccnt(0);
#else
  if (keep3) asm volatile("s_wait_asynccnt 3" ::: "memory");
  else       asm volatile("s_wait_asynccnt 0" ::: "memory");
#endif
}

// ---------------------------------------------------------------------------
// Kernel B (defined first so its disassembly leads the .s file):
// edge scatter, out[row] += alpha*w*x[col] via fp32 HW atomics.
// Edge tuples stream through LDS with async-to-LDS double buffering.
// One wave32 per edge; lane handles 4 consecutive features (float4).
// ---------------------------------------------------------------------------
__global__ void spmm_scatter_kernel(const float* __restrict__ x,
                                    const int* __restrict__ eidx,   // rows[E], cols[E]
                                    const float* __restrict__ ew,
                                    const float* __restrict__ alphaT,
                                    float* __restrict__ out,
                                    int E, int numTiles) {
  __shared__ int   sRow[2][TILE];
  __shared__ int   sCol[2][TILE];
  __shared__ float sW[2][TILE];

  const int tid      = threadIdx.x;
  const int lane     = tid & 31;
  const int waveBase = tid & ~31;  // wave_id * 32
  const float alpha  = 0.1f / (1.0f + __expf(-alphaT[0]));

  const int* __restrict__ rowg = eidx;
  const int* __restrict__ colg = eidx + E;

  const int t0 = blockIdx.x;
  if (t0 >= numTiles) return;

  // stage one edge tuple per thread; clamp so every wave always issues
  // exactly 3 async ops (keeps ASYNCcnt bookkeeping uniform)
  auto stage = [&](int tileStart, int buf) {
    int e = tileStart + tid;
    if (e >= E) e = E - 1;
#if HAS_ASYNC_LDS
    async_copy_b32(rowg + e, &sRow[buf][tid]);
    async_copy_b32(colg + e, &sCol[buf][tid]);
    async_copy_b32(ew + e,   &sW[buf][tid]);
#else
    sRow[buf][tid] = rowg[e];
    sCol[buf][tid] = colg[e];
    sW[buf][tid]   = ew[e];
#endif
  };

  int buf = 0;
  stage(t0 * TILE, buf);

  for (int t = t0; t < numTiles; t += gridDim.x) {
    const int nt = t + gridDim.x;
    if (nt < numTiles) {
      stage(nt * TILE, buf ^ 1);  // prefetch next tile (stays in flight)
      wait_async_le(1);           // retire only current tile's 3 async ops
    } else {
      wait_async_le(0);
    }
    __syncthreads();

    const int tileStart = t * TILE;
    // only the final tile is partial: hoist the bound so the inner loop can
    // be unrolled / software-pipelined (atomics are fire-and-forget)
    int jEnd = E - tileStart - waveBase;
    if (jEnd > 32) jEnd = 32;
#pragma unroll 4
    for (int j = 0; j < jEnd; ++j) {
      // uniform LDS read -> force scalar for SALU base-address computation
      const int row = __builtin_amdgcn_readfirstlane(sRow[buf][waveBase + j]);
      const int col = __builtin_amdgcn_readfirstlane(sCol[buf][waveBase + j]);
      const float s = alpha * sW[buf][waveBase + j];

      const float4 xv =
          *reinterpret_cast<const float4*>(x + (size_t)col * DFEAT + lane * 4);
      float* op = out + (size_t)row * DFEAT + lane * 4;
      unsafeAtomicAdd(op + 0, s * xv.x);
      unsafeAtomicAdd(op + 1, s * xv.y);
      unsafeAtomicAdd(op + 2, s * xv.z);
      unsafeAtomicAdd(op + 3, s * xv.w);
    }
    __syncthreads();  // tile fully consumed before its buffer is re-staged
    buf ^= 1;
  }
}

// ---------------------------------------------------------------------------
// Kernel A: elementwise part, float4 vectorized. Fully overwrites d_out.
// ---------------------------------------------------------------------------
__global__ void ode_elemwise_kernel(const float* __restrict__ x,
                                    const float* __restrict__ x0,
                                    const float* __restrict__ alphaT,
                                    const float* __restrict__ betaT,
                                    const float* __restrict__ srcT,
                                    float* __restrict__ out, int total4) {
  const float alpha = 0.1f / (1.0f + __expf(-alphaT[0]));
  const float beta  = 0.1f / (1.0f + __expf(-betaT[0]));
  const float src   = srcT[0];
  int i = blockIdx.x * blockDim.x + threadIdx.x;
  const int stride = gridDim.x * blockDim.x;
  for (; i < total4; i += stride) {
    const float4 xv  = reinterpret_cast<const float4*>(x)[i];
    const float4 x0v = reinterpret_cast<const float4*>(x0)[i];
    float4 o;
    o.x = -alpha * xv.x + beta * (xv.x - xv.x * xv.x) +
          0.1f * fminf(1.0f, fmaxf(-1.0f, src * x0v.x));
    o.y = -alpha * xv.y + beta * (xv.y - xv.y * xv.y) +
          0.1f * fminf(1.0f, fmaxf(-1.0f, src * x0v.y));
    o.z = -alpha * xv.z + beta * (xv.z - xv.z * xv.z) +
          0.1f * fminf(1.0f, fmaxf(-1.0f, src * x0v.z));
    o.w = -alpha * xv.w + beta * (xv.w - xv.w * xv.w) +
          0.1f * fminf(1.0f, fmaxf(-1.0f, src * x0v.w));
    reinterpret_cast<float4*>(out)[i] = o;
  }
}

// ---------------------------------------------------------------------------
extern "C" void kernel_launch(void* const* d_in, const int* in_sizes, int n_in,
                              void* d_out, int out_size, void* d_ws, size_t ws_size,
                              hipStream_t stream) {
  // setup_inputs order: t, x, edge_index, edge_weight, x0, alpha, beta, source
  const float* x    = (const float*)d_in[1];
  const int*   eidx = (const int*)d_in[2];   // harness: integer -> const int*
  const float* ew   = (const float*)d_in[3];
  const float* x0   = (const float*)d_in[4];
  const float* aT   = (const float*)d_in[5];
  const float* bT   = (const float*)d_in[6];
  const float* sT   = (const float*)d_in[7];
  float* out = (float*)d_out;

  const int ND = in_sizes[1];                 // N*D
  const int E  = in_sizes[3];

  const int total4 = ND / 4;
  int gridA = (total4 + 255) / 256;
  ode_elemwise_kernel<<<gridA, 256, 0, stream>>>(x, x0, aT, bT, sT, out, total4);

  const int numTiles = (E + TILE - 1) / TILE;
  int gridB = numTiles < 2048 ? numTiles : 2048;
  spmm_scatter_kernel<<<gridB, 256, 0, stream>>>(x, eidx, ew, aT, out, E, numTiles);
}